// SEBlock_65343632441501
// MI455X (gfx1250) — compile-verified
//
#include <hip/hip_runtime.h>
#include <math.h>

typedef float v2f __attribute__((ext_vector_type(2)));
typedef float v8f __attribute__((ext_vector_type(8)));

#define N_SAMP 32
#define CCH    512
#define HH     56
#define WW     56
#define PLANE  (HH*WW)          // 3136 floats
#define PLANE4 (PLANE/4)        // 784 float4
#define K1     8192             // C*4*4
#define K2     512
#define TOPK   102              // int(0.2*512)

// ---------------------------------------------------------------------------
// Pass 1: adaptive avg-pool 4x4 (14x14 windows) + sigmoid.  One block/plane.
// Deterministic: fixed-order partial sums (row-chunk -> bin), no atomics.
// ---------------------------------------------------------------------------
__global__ void pool_sigmoid_kernel(const float* __restrict__ X,
                                    float* __restrict__ P) {
  __shared__ float rowsum[224];               // [row(56)][jbin(4)]
  const int plane = blockIdx.x;               // n*C + c
  const float* xp = X + (size_t)plane * PLANE;
  const int t = threadIdx.x;                  // 256 threads
  if (t < 224) {
    const int row = t >> 2;
    const int jb  = t & 3;
    const float* rp = xp + row * WW + jb * 14;
    float a = 0.f;
#pragma unroll
    for (int e = 0; e < 14; ++e) a += rp[e];
    rowsum[t] = a;
  }
  __syncthreads();
  if (t < 16) {
    const int bi = t >> 2, bj = t & 3;
    float a = 0.f;
#pragma unroll
    for (int r = 0; r < 14; ++r) a += rowsum[(bi * 14 + r) * 4 + bj];
    const float m = a * (1.0f / 196.0f);
    P[(size_t)plane * 16 + t] = 1.0f / (1.0f + expf(-m));
  }
}

// ---------------------------------------------------------------------------
// Spectral-norm helpers: t = W^T u  (coalesced column mat-vec)
// ---------------------------------------------------------------------------
__global__ void matvec_cols_kernel(const float* __restrict__ W,
                                   const float* __restrict__ u,
                                   float* __restrict__ t,
                                   int rows, int K) {
  const int j = blockIdx.x * blockDim.x + threadIdx.x;
  if (j >= K) return;
  float acc = 0.f;
  for (int i = 0; i < rows; ++i) acc += W[(size_t)i * K + j] * u[i];
  t[j] = acc;
}

// y[row] = dot(W[row,:], t)  — block-per-row tree reduction (deterministic)
__global__ void matvec_rows_kernel(const float* __restrict__ W,
                                   const float* __restrict__ t,
                                   float* __restrict__ y, int K) {
  __shared__ float red[256];
  const int row = blockIdx.x;
  const float* wr = W + (size_t)row * K;
  float acc = 0.f;
  for (int j = threadIdx.x; j < K; j += 256) acc += wr[j] * t[j];
  red[threadIdx.x] = acc;
  __syncthreads();
  for (int s = 128; s > 0; s >>= 1) {
    if (threadIdx.x < s) red[threadIdx.x] += red[threadIdx.x + s];
    __syncthreads();
  }
  if (threadIdx.x == 0) y[row] = red[0];
}

// single-block sum of squares -> *outp
__global__ void reduce_sumsq_kernel(const float* __restrict__ v, int n,
                                    float* __restrict__ outp) {
  __shared__ float red[256];
  float acc = 0.f;
  for (int i = threadIdx.x; i < n; i += 256) { float x = v[i]; acc += x * x; }
  red[threadIdx.x] = acc;
  __syncthreads();
  for (int s = 128; s > 0; s >>= 1) {
    if (threadIdx.x < s) red[threadIdx.x] += red[threadIdx.x + s];
    __syncthreads();
  }
  if (threadIdx.x == 0) *outp = red[0];
}

// ---------------------------------------------------------------------------
// WMMA fp32 GEMM:  D(32 x 512) = A(32 x K) * B(512 x K)^T   via 16x16x4 f32.
// One wave (32 threads) per 16x16 output tile; K swept 4 at a time.
// A-frag per ISA 7.12.2: lanes 0-15 carry {K,K+1}, lanes 16-31 carry {K+2,K+3}
// for rows M=lane%16 (B mirrors with N=lane%16).  ss = {sumsq_t, sumsq_Wt}.
// ---------------------------------------------------------------------------
__global__ void gemm_wmma_relu_kernel(const float* __restrict__ A,
                                      const float* __restrict__ B,
                                      const float* __restrict__ ss,
                                      float* __restrict__ D) {
  const int tile = blockIdx.x;          // 64 tiles: 2 (M) x 32 (N)
  const int mt = tile & 1;
  const int nt = tile >> 1;
  const int lane = threadIdx.x;
  const int hi = lane >> 4;             // 0: K{0,1}, 1: K{2,3}
  const int lo = lane & 15;
  const float* arow = A + (size_t)(mt * 16 + lo) * K1;
  const float* brow = B + (size_t)(nt * 16 + lo) * K1;
  v8f c = {};
#pragma unroll 4
  for (int k = 0; k < K1; k += 4) {
    const int k0 = k + (hi << 1);
    v2f a = *(const v2f*)(arow + k0);
    v2f b = *(const v2f*)(brow + k0);
    c = __builtin_amdgcn_wmma_f32_16x16x4_f32(false, a, false, b,
                                              (short)0, c, false, false);
  }
  const float sigma = sqrtf(ss[1]) / (sqrtf(ss[0]) + 1e-12f);
  const float inv = 1.0f / sigma;
#pragma unroll
  for (int r = 0; r < 8; ++r) {
    float v = c[r] * inv;
    v = v > 0.f ? v : 0.f;                       // relu
    D[(size_t)(mt * 16 + r + hi * 8) * CCH + nt * 16 + lo] = v;
  }
}

__global__ void gemm_wmma_sigmoid_kernel(const float* __restrict__ A,
                                         const float* __restrict__ B,
                                         const float* __restrict__ ss,
                                         float* __restrict__ D) {
  const int tile = blockIdx.x;
  const int mt = tile & 1;
  const int nt = tile >> 1;
  const int lane = threadIdx.x;
  const int hi = lane >> 4;
  const int lo = lane & 15;
  const float* arow = A + (size_t)(mt * 16 + lo) * K2;
  const float* brow = B + (size_t)(nt * 16 + lo) * K2;
  v8f c = {};
#pragma unroll 4
  for (int k = 0; k < K2; k += 4) {
    const int k0 = k + (hi << 1);
    v2f a = *(const v2f*)(arow + k0);
    v2f b = *(const v2f*)(brow + k0);
    c = __builtin_amdgcn_wmma_f32_16x16x4_f32(false, a, false, b,
                                              (short)0, c, false, false);
  }
  const float sigma = sqrtf(ss[1]) / (sqrtf(ss[0]) + 1e-12f);
  const float inv = 1.0f / sigma;
#pragma unroll
  for (int r = 0; r < 8; ++r) {
    const float v = 1.0f / (1.0f + expf(-c[r] * inv));   // sigmoid
    D[(size_t)(mt * 16 + r + hi * 8) * CCH + nt * 16 + lo] = v;
  }
}

// ---------------------------------------------------------------------------
// Exact top-k mask per row (k=102), ties -> lower index (matches lax.top_k).
// ---------------------------------------------------------------------------
__global__ void topk_mask_kernel(const float* __restrict__ S,
                                 float* __restrict__ Wgt) {
  __shared__ float sh[CCH];
  const int n = blockIdx.x;
  const int cidx = threadIdx.x;          // 512 threads
  const float v = S[(size_t)n * CCH + cidx];
  sh[cidx] = v;
  __syncthreads();
  int rank = 0;
  for (int j = 0; j < CCH; ++j) {
    const float sj = sh[j];
    rank += (sj > v) || (sj == v && j < cidx);
  }
  Wgt[(size_t)n * CCH + cidx] = (rank < TOPK) ? v : 0.0f;
}

// ---------------------------------------------------------------------------
// Pass 2: out = x * weight[plane], float4 streaming.  One block per plane.
// ---------------------------------------------------------------------------
__global__ void scale_kernel(const float* __restrict__ X,
                             const float* __restrict__ Wgt,
                             float* __restrict__ O) {
  const int plane = blockIdx.x;
  const float w = Wgt[plane];
  const float4* xs = (const float4*)(X + (size_t)plane * PLANE);
  float4* os = (float4*)(O + (size_t)plane * PLANE);
  for (int i = threadIdx.x; i < PLANE4; i += blockDim.x) {
    float4 v = xs[i];
    v.x *= w; v.y *= w; v.z *= w; v.w *= w;
    os[i] = v;
  }
}

// ---------------------------------------------------------------------------
extern "C" void kernel_launch(void* const* d_in, const int* in_sizes, int n_in,
                              void* d_out, int out_size, void* d_ws, size_t ws_size,
                              hipStream_t stream) {
  (void)in_sizes; (void)n_in; (void)out_size; (void)ws_size;
  const float* x  = (const float*)d_in[0];
  const float* w1 = (const float*)d_in[1];   // (512,512,4,4) -> rows of 8192
  const float* w2 = (const float*)d_in[2];   // (512,512,1,1) -> rows of 512
  const float* u1 = (const float*)d_in[3];
  const float* u2 = (const float*)d_in[4];
  float* out = (float*)d_out;

  float* ws  = (float*)d_ws;
  float* ss1 = ws;               // {sumsq_t1, sumsq_y1}
  float* ss2 = ws + 2;           // {sumsq_t2, sumsq_y2}
  float* t1  = ws + 16;          // 8192
  float* t2  = t1 + K1;          // 512
  float* y1  = t2 + K2;          // 512
  float* y2  = y1 + CCH;         // 512
  float* p   = ws + 16384;       // 32*512*16 = 262144
  float* h   = p + N_SAMP * K1;  // 16384
  float* s   = h + N_SAMP * CCH; // 16384
  float* wgt = s + N_SAMP * CCH; // 16384

  const int planes = N_SAMP * CCH;   // 16384

  // Pass 1: pool + sigmoid (dominant HBM read #1)
  pool_sigmoid_kernel<<<planes, 256, 0, stream>>>(x, p);

  // Spectral norms (tiny)
  matvec_cols_kernel<<<K1 / 256, 256, 0, stream>>>(w1, u1, t1, CCH, K1);
  reduce_sumsq_kernel<<<1, 256, 0, stream>>>(t1, K1, &ss1[0]);
  matvec_rows_kernel<<<CCH, 256, 0, stream>>>(w1, t1, y1, K1);
  reduce_sumsq_kernel<<<1, 256, 0, stream>>>(y1, CCH, &ss1[1]);

  matvec_cols_kernel<<<K2 / 256, 256, 0, stream>>>(w2, u2, t2, CCH, K2);
  reduce_sumsq_kernel<<<1, 256, 0, stream>>>(t2, K2, &ss2[0]);
  matvec_rows_kernel<<<CCH, 256, 0, stream>>>(w2, t2, y2, K2);
  reduce_sumsq_kernel<<<1, 256, 0, stream>>>(y2, CCH, &ss2[1]);

  // Channel-mixing GEMMs on the matrix pipe (fp32 WMMA 16x16x4)
  gemm_wmma_relu_kernel<<<64, 32, 0, stream>>>(p, w1, ss1, h);
  gemm_wmma_sigmoid_kernel<<<64, 32, 0, stream>>>(h, w2, ss2, s);

  // Exact top-k gating
  topk_mask_kernel<<<N_SAMP, CCH, 0, stream>>>(s, wgt);

  // Pass 2: broadcast scale (dominant HBM read #2 + write)
  scale_kernel<<<planes, 256, 0, stream>>>(x, wgt, out);
}